// DenoisingSharpening_44942537786198
// MI455X (gfx1250) — compile-verified
//
#include <hip/hip_runtime.h>

#define THREADS 256
#define IMG_FLOATS 12288   // 64*64*3 floats = 48 KB per image
#define PPT 16             // pixels per thread (4096 / 256)

typedef int v4i __attribute__((vector_size(16)));

#if defined(__HIP_DEVICE_COMPILE__) && \
    __has_builtin(__builtin_amdgcn_global_load_async_to_lds_b128) && \
    __has_builtin(__builtin_amdgcn_s_wait_asynccnt)
#define USE_ASYNC_LDS 1
typedef __attribute__((address_space(3))) v4i* as3_v4i_p;
#elif defined(__HIP_DEVICE_COMPILE__)
#warning "gfx1250 async-to-LDS builtins unavailable; using VALU copy fallback"
#endif

__device__ __forceinline__ float clamp01(float v) {
    return fminf(fmaxf(v, 1e-5f), 1.0f);
}

// Streaming (non-temporal) store: output is write-once, never re-read.
__device__ __forceinline__ void nt_store(float* p, float v) {
    __builtin_nontemporal_store(v, p);
}

// Per-channel tail: detail/noise masks + sharpening. Accumulates the two
// image-wide reduction partials.
__device__ __forceinline__ float finish_channel(float xc, float bf, float gf,
                                                float gain, float offset,
                                                float inv_tau, float lam,
                                                float kpos,
                                                float& nsum, float& dsum) {
    float detail = xc - gf;
    float ad = fabsf(detail);
    float ne = fminf(ad * gain / fmaxf(xc + offset, 1e-5f), 10.0f);
    nsum += ne;
    dsum += ad;
    float q = ne * inv_tau;
    float nm = 1.0f - __expf(fmaxf(-q * q, -88.0f));
    nm *= nm;
    float dm = 1.0f / (1.0f + __expf(-kpos * (ad - 0.002f)));
    return clamp01(bf + lam * detail * nm * dm);
}

__global__ __launch_bounds__(THREADS)
void DenoisingSharpening_44942537786198_kernel(const float* __restrict__ images,
                                               const float* __restrict__ params,
                                               const float* __restrict__ kscalar,
                                               float* __restrict__ out) {
    __shared__ float s_img[IMG_FLOATS];     // one full 64x64x3 image, HWC
    __shared__ float s_red[2 * THREADS];    // noise / detail partial sums
    __shared__ int   s_skip;

    const int img = blockIdx.x;
    const int tid = threadIdx.x;
    const float* gimg = images + (size_t)img * IMG_FLOATS;
    float* gout = out + (size_t)img * IMG_FLOATS;

    // ---- Stage image into LDS (async copy on CDNA5, ASYNCcnt-tracked) ----
#ifdef USE_ASYNC_LDS
    {
        const v4i* gsrc = (const v4i*)gimg;   // generic ptr, 16B elements
        v4i* ldst = (v4i*)s_img;              // generic ptr into LDS object
#pragma unroll
        for (int i = 0; i < 12; ++i) {
            int e = i * THREADS + tid;        // 16B element index
            __builtin_amdgcn_global_load_async_to_lds_b128(
                (v4i*)(gsrc + e), (as3_v4i_p)(ldst + e), 0, 0);
        }
        __builtin_amdgcn_s_wait_asynccnt(0);
    }
#else
#pragma unroll
    for (int i = 0; i < 12; ++i) {
        int f = (i * THREADS + tid) * 4;
        float4 v = *(const float4*)(gimg + f);
        *(float4*)(s_img + f) = v;
    }
#endif

    // ---- Per-image parameters (uniform across block; SMEM/L2-cached) ----
    const float* pp = params + img * 7;
    float sigma_s = fminf(fmaxf(pp[0], 0.2f), 5.0f);
    float sigma_r = fminf(fmaxf(pp[1], 0.01f), 1.0f);
    float sigma_f = fminf(fmaxf(pp[2], 0.2f), 3.0f);
    float lam     = fminf(fmaxf(pp[3], 0.1f), 2.0f);
    float tau     = fminf(fmaxf(pp[4], 0.5f), 5.0f);
    float gain    = fminf(fmaxf(pp[5], 0.2f), 2.0f);
    float offset  = fminf(fmaxf(pp[6], 0.01f), 1.0f);
    float kpos    = fmaxf(fabsf(kscalar[0]), 1.0f);
    float inv2s2  = -0.5f / (sigma_r * sigma_r);
    float inv_tau = 1.0f / tau;

    // Normalized 1D Gaussians over taps [-1,0,1]: g = [a, b, a]
    float es = __expf(-0.5f / (sigma_s * sigma_s));
    float sn = 1.0f / (1.0f + 2.0f * es);
    float gs[3] = {es * sn, sn, es * sn};     // spatial (bilateral)
    float ef = __expf(-0.5f / (sigma_f * sigma_f));
    float fn = 1.0f / (1.0f + 2.0f * ef);
    float gw[3] = {ef * fn, fn, ef * fn};     // gaussian (detail base)

    __syncthreads();

    float nsum = 0.0f, dsum = 0.0f;

#pragma unroll 1
    for (int it = 0; it < PPT; ++it) {
        int p = tid + it * THREADS;           // 0..4095, coalesced across wave
        int h = p >> 6, w = p & 63;
        int c0 = p * 3;
        float xc0 = s_img[c0], xc1 = s_img[c0 + 1], xc2 = s_img[c0 + 2];

        // reflect padding (numpy 'reflect': -1 -> 1, 64 -> 62)
        int rows[3] = {((h == 0) ? 1 : h - 1) * 64, h * 64,
                       ((h == 63) ? 62 : h + 1) * 64};
        int cols[3] = {(w == 0) ? 1 : w - 1, w, (w == 63) ? 62 : w + 1};

        float wsum = 0.0f;
        float b0 = 0.0f, b1 = 0.0f, b2 = 0.0f;   // bilateral accum
        float g0 = 0.0f, g1 = 0.0f, g2 = 0.0f;   // gaussian accum
#pragma unroll
        for (int i = 0; i < 3; ++i) {
#pragma unroll
            for (int j = 0; j < 3; ++j) {
                int idx = (rows[i] + cols[j]) * 3;
                float n0 = s_img[idx], n1 = s_img[idx + 1], n2 = s_img[idx + 2];
                float d0 = n0 - xc0, d1 = n1 - xc1, d2 = n2 - xc2;
                float ck = __expf(inv2s2 * (d0 * d0 + d1 * d1 + d2 * d2));
                float wk = gs[i] * gs[j] * ck;
                wsum += wk;
                b0 = fmaf(n0, wk, b0);
                b1 = fmaf(n1, wk, b1);
                b2 = fmaf(n2, wk, b2);
                float wg = gw[i] * gw[j];
                g0 = fmaf(n0, wg, g0);
                g1 = fmaf(n1, wg, g1);
                g2 = fmaf(n2, wg, g2);
            }
        }
        // wsum >= center weight b^2 * exp(0) > 0.1, so bf is always finite
        // (reference's nan_to_num is a no-op).
        float rw = 1.0f / wsum;
        float r0 = finish_channel(xc0, b0 * rw, g0, gain, offset, inv_tau, lam, kpos, nsum, dsum);
        float r1 = finish_channel(xc1, b1 * rw, g1, gain, offset, inv_tau, lam, kpos, nsum, dsum);
        float r2 = finish_channel(xc2, b2 * rw, g2, gain, offset, inv_tau, lam, kpos, nsum, dsum);
        nt_store(gout + c0,     r0);
        nt_store(gout + c0 + 1, r1);
        nt_store(gout + c0 + 2, r2);
    }

    // ---- Image-wide reduction for should_skip ----
    s_red[tid] = nsum;
    s_red[THREADS + tid] = dsum;
    __syncthreads();
    for (int off = THREADS / 2; off > 0; off >>= 1) {
        if (tid < off) {
            s_red[tid] += s_red[tid + off];
            s_red[THREADS + tid] += s_red[THREADS + tid + off];
        }
        __syncthreads();
    }
    if (tid == 0) {
        const float inv_n = 1.0f / (float)IMG_FLOATS;
        float avg_noise = s_red[0] * inv_n;
        float avg_detail = s_red[THREADS] * inv_n;
        s_skip = (avg_noise < 1e-4f) || (avg_detail < 1e-4f) ? 1 : 0;
    }
    __syncthreads();

    // Rare path: whole image passes through unchanged (clipped). Same thread
    // rewrites the same addresses, so store ordering is guaranteed.
    if (s_skip) {
#pragma unroll 1
        for (int it = 0; it < PPT; ++it) {
            int c0 = (tid + it * THREADS) * 3;
            nt_store(gout + c0,     clamp01(s_img[c0]));
            nt_store(gout + c0 + 1, clamp01(s_img[c0 + 1]));
            nt_store(gout + c0 + 2, clamp01(s_img[c0 + 2]));
        }
    }
}

extern "C" void kernel_launch(void* const* d_in, const int* in_sizes, int n_in,
                              void* d_out, int out_size, void* d_ws, size_t ws_size,
                              hipStream_t stream) {
    (void)n_in; (void)out_size; (void)d_ws; (void)ws_size;
    const float* images = (const float*)d_in[0];
    const float* params = (const float*)d_in[1];
    const float* k      = (const float*)d_in[2];
    float* out = (float*)d_out;

    int n_img = in_sizes[1] / 7;  // (B*P) = 512 images
    DenoisingSharpening_44942537786198_kernel<<<n_img, THREADS, 0, stream>>>(
        images, params, k, out);
}